// ABCLinear_52682068853276
// MI455X (gfx1250) — compile-verified
//
#include <hip/hip_runtime.h>

typedef __attribute__((ext_vector_type(2))) float v2f;
typedef __attribute__((ext_vector_type(8))) float v8f;
typedef __attribute__((ext_vector_type(4))) int   v4i;

#define TOKENS 8192
#define D_IN   4096
#define D_OUT  4096
#define RANK   64
#define NF     2

// Async global->LDS staging (CDNA5 GLOBAL_LOAD_ASYNC_TO_LDS_B128, ASYNCcnt).
#if defined(__has_builtin)
#if __has_builtin(__builtin_amdgcn_global_load_async_to_lds_b128) && \
    __has_builtin(__builtin_amdgcn_s_wait_asynccnt)
#define USE_ASYNC_LDS 1
#endif
#endif
#ifndef USE_ASYNC_LDS
#define USE_ASYNC_LDS 0
#endif

#if USE_ASYNC_LDS
__device__ __forceinline__ void async_cp16(const float* gsrc, float* ldst) {
  // builtin expects non-const generic pointers to 16B int vectors
  __builtin_amdgcn_global_load_async_to_lds_b128((v4i*)gsrc, (v4i*)ldst, 0, 0);
}
#endif

// ---------------------------------------------------------------------------
// Kernel 1: AB[f][o][k] = sum_r tanh(As[f][o][r]) * Bs[f][r][k]
// 2*4096*64 elements, ~67 MFLOP -> plain VALU, negligible cost.
// ---------------------------------------------------------------------------
__global__ __launch_bounds__(256) void ab_kernel(const float* __restrict__ As,
                                                 const float* __restrict__ Bs,
                                                 float* __restrict__ AB) {
  int idx = blockIdx.x * 256 + threadIdx.x;          // [0, NF*D_OUT*RANK)
  int f   = idx / (D_OUT * RANK);
  int rem = idx - f * (D_OUT * RANK);
  int o   = rem / RANK;
  int k   = rem - o * RANK;
  const float* a = As + (size_t)f * D_OUT * RANK + (size_t)o * RANK;
  const float* b = Bs + (size_t)f * RANK * RANK + k;
  float acc = 0.f;
#pragma unroll 8
  for (int r = 0; r < RANK; ++r) acc += tanhf(a[r]) * b[r * RANK];
  AB[idx] = acc;
}

// ---------------------------------------------------------------------------
// Kernel 2: Weff[o][i] = W[o][i] + sum_f sum_k AB[f][o][k] * Cs[f][k][i]
// One wave per 16x16 tile, WMMA f32 16x16x4 over K = 2*64 = 128.
// AB (2MB) + Cs (2MB) are L2-resident; perf irrelevant (4.3 GFLOP).
// ---------------------------------------------------------------------------
__global__ __launch_bounds__(256) void weff_kernel(const float* __restrict__ AB,
                                                   const float* __restrict__ Cs,
                                                   const float* __restrict__ W,
                                                   float* __restrict__ Weff) {
  const int lane = threadIdx.x & 31;
  const int wave = threadIdx.x >> 5;
  const int tile = blockIdx.x * 8 + wave;
  const int tilesN = D_IN / 16;                      // 256
  const int tm = tile / tilesN;
  const int tn = tile - tm * tilesN;
  const int o0 = tm * 16, n0 = tn * 16;
  const int hl = lane & 15, hi = lane >> 4;

  v8f acc = {};
  for (int f = 0; f < NF; ++f) {
    const float* abf = AB + (size_t)f * D_OUT * RANK;
    const float* cf  = Cs + (size_t)f * RANK * D_IN;
#pragma unroll
    for (int kk = 0; kk < RANK / 4; ++kk) {
      const int k = kk * 4 + hi * 2;                 // lane-half K pair
      v2f a = *(const v2f*)(abf + (size_t)(o0 + hl) * RANK + k);
      v2f b;
      b.x = cf[(size_t)k * D_IN + n0 + hl];
      b.y = cf[(size_t)(k + 1) * D_IN + n0 + hl];
      acc = __builtin_amdgcn_wmma_f32_16x16x4_f32(false, a, false, b,
                                                  (short)0, acc, false, false);
    }
  }
  const int row0 = o0 + hi * 8;
  const int col  = n0 + hl;
#pragma unroll
  for (int r = 0; r < 8; ++r) {
    size_t off = (size_t)(row0 + r) * D_IN + col;
    Weff[off] = W[off] + acc[r];
  }
}

// ---------------------------------------------------------------------------
// Kernel 3: Out[t][o] = sum_i x[t][i] * Weff[o][i] + bias[o]
// M=8192 N=4096 K=4096 fp32, compute-bound (274 GFLOP vs ~14us of HBM).
// 256-thread block (8 wave32), 128x128x16 block tile, double-buffered LDS,
// async global->LDS staging when available. Waves 4(M) x 2(N); each wave a
// 32x64 tile = 2x4 v_wmma_f32_16x16x4_f32 accumulators.
// LDS rows padded to 20 floats: 80B rows keep 16B stores aligned and make the
// 16-lane b64 fragment reads bank-conflict-free (20*L mod 64 distinct).
// ---------------------------------------------------------------------------
#define BM   128
#define BN   128
#define BK   16
#define LDSW 20

__global__ __launch_bounds__(256) void gemm_kernel(const float* __restrict__ X,
                                                   const float* __restrict__ Weff,
                                                   const float* __restrict__ bias,
                                                   float* __restrict__ Out) {
  __shared__ float sA[2][BM * LDSW];   // 10240 B per buffer
  __shared__ float sB[2][BN * LDSW];   // total 40 KB

  const int tid   = threadIdx.x;
  const int lane  = tid & 31;
  const int wave  = tid >> 5;
  const int hl    = lane & 15, hi = lane >> 4;
  const int waveM = (wave & 3) * 32;   // 4 waves along M
  const int waveN = (wave >> 2) * 64;  // 2 waves along N
  const int m0 = blockIdx.y * BM;
  const int n0 = blockIdx.x * BN;
  const int K  = D_IN;
  const int stages = K / BK;           // 256

  // per-thread staging coordinates: 512 float4 per tile, 2 per thread
  int lrow[2], lcol[2];
#pragma unroll
  for (int j = 0; j < 2; ++j) {
    int l = j * 256 + tid;
    lrow[j] = l >> 2;
    lcol[j] = (l & 3) * 4;
  }

  v8f acc[2][4];
#pragma unroll
  for (int i = 0; i < 2; ++i)
#pragma unroll
    for (int j = 0; j < 4; ++j) acc[i][j] = (v8f){};

#if USE_ASYNC_LDS
  // ---- async pipeline: stage t lands in lds[t & 1] ----
#pragma unroll
  for (int j = 0; j < 2; ++j) {
    async_cp16(X    + (size_t)(m0 + lrow[j]) * K + lcol[j],
               &sA[0][lrow[j] * LDSW + lcol[j]]);
    async_cp16(Weff + (size_t)(n0 + lrow[j]) * K + lcol[j],
               &sB[0][lrow[j] * LDSW + lcol[j]]);
  }
  for (int s = 0; s < stages; ++s) {
    const int buf = s & 1;
    if (s + 1 < stages) {
      const int kn = (s + 1) * BK;
      const int nbuf = buf ^ 1;
#pragma unroll
      for (int j = 0; j < 2; ++j) {
        async_cp16(X    + (size_t)(m0 + lrow[j]) * K + kn + lcol[j],
                   &sA[nbuf][lrow[j] * LDSW + lcol[j]]);
        async_cp16(Weff + (size_t)(n0 + lrow[j]) * K + kn + lcol[j],
                   &sB[nbuf][lrow[j] * LDSW + lcol[j]]);
      }
      __builtin_amdgcn_s_wait_asynccnt(4);  // stage-s loads (prev iter) done
    } else {
      __builtin_amdgcn_s_wait_asynccnt(0);  // nothing else in flight
    }
    __syncthreads();                        // all waves' stage-s loads done
#pragma unroll
    for (int kk = 0; kk < BK / 4; ++kk) {
      const int kl = kk * 4 + hi * 2;
      v2f af[2], bf[4];
#pragma unroll
      for (int mf = 0; mf < 2; ++mf)
        af[mf] = *(const v2f*)(&sA[buf][(waveM + mf * 16 + hl) * LDSW + kl]);
#pragma unroll
      for (int nf = 0; nf < 4; ++nf)
        bf[nf] = *(const v2f*)(&sB[buf][(waveN + nf * 16 + hl) * LDSW + kl]);
#pragma unroll
      for (int mf = 0; mf < 2; ++mf)
#pragma unroll
        for (int nf = 0; nf < 4; ++nf)
          acc[mf][nf] = __builtin_amdgcn_wmma_f32_16x16x4_f32(
              false, af[mf], false, bf[nf], (short)0, acc[mf][nf], false, false);
    }
    __syncthreads();  // readers of lds[buf] done before stage s+2 overwrites it
  }
#else
  // ---- fallback: register-staged double buffering ----
  float4 ra[2], rb[2];
#pragma unroll
  for (int j = 0; j < 2; ++j) {
    ra[j] = *(const float4*)(X    + (size_t)(m0 + lrow[j]) * K + lcol[j]);
    rb[j] = *(const float4*)(Weff + (size_t)(n0 + lrow[j]) * K + lcol[j]);
  }
#pragma unroll
  for (int j = 0; j < 2; ++j) {
    *(float4*)(&sA[0][lrow[j] * LDSW + lcol[j]]) = ra[j];
    *(float4*)(&sB[0][lrow[j] * LDSW + lcol[j]]) = rb[j];
  }
  for (int s = 0; s < stages; ++s) {
    __syncthreads();
    const int buf = s & 1;
    if (s + 1 < stages) {
      const int kn = (s + 1) * BK;
#pragma unroll
      for (int j = 0; j < 2; ++j) {
        ra[j] = *(const float4*)(X    + (size_t)(m0 + lrow[j]) * K + kn + lcol[j]);
        rb[j] = *(const float4*)(Weff + (size_t)(n0 + lrow[j]) * K + kn + lcol[j]);
      }
    }
#pragma unroll
    for (int kk = 0; kk < BK / 4; ++kk) {
      const int kl = kk * 4 + hi * 2;
      v2f af[2], bf[4];
#pragma unroll
      for (int mf = 0; mf < 2; ++mf)
        af[mf] = *(const v2f*)(&sA[buf][(waveM + mf * 16 + hl) * LDSW + kl]);
#pragma unroll
      for (int nf = 0; nf < 4; ++nf)
        bf[nf] = *(const v2f*)(&sB[buf][(waveN + nf * 16 + hl) * LDSW + kl]);
#pragma unroll
      for (int mf = 0; mf < 2; ++mf)
#pragma unroll
        for (int nf = 0; nf < 4; ++nf)
          acc[mf][nf] = __builtin_amdgcn_wmma_f32_16x16x4_f32(
              false, af[mf], false, bf[nf], (short)0, acc[mf][nf], false, false);
    }
    if (s + 1 < stages) {
      const int nbuf = (s + 1) & 1;
#pragma unroll
      for (int j = 0; j < 2; ++j) {
        *(float4*)(&sA[nbuf][lrow[j] * LDSW + lcol[j]]) = ra[j];
        *(float4*)(&sB[nbuf][lrow[j] * LDSW + lcol[j]]) = rb[j];
      }
    }
  }
#endif

  // Epilogue: D frag vgpr r -> row (base + hi*8 + r), col (base + hl); add bias.
#pragma unroll
  for (int nf = 0; nf < 4; ++nf) {
    const int col = n0 + waveN + nf * 16 + hl;
    const float bv = bias[col];
#pragma unroll
    for (int mf = 0; mf < 2; ++mf) {
      const int rowb = m0 + waveM + mf * 16 + hi * 8;
#pragma unroll
      for (int r = 0; r < 8; ++r)
        Out[(size_t)(rowb + r) * D_OUT + col] = acc[mf][nf][r] + bv;
    }
  }
}

// ---------------------------------------------------------------------------
extern "C" void kernel_launch(void* const* d_in, const int* in_sizes, int n_in,
                              void* d_out, int out_size, void* d_ws, size_t ws_size,
                              hipStream_t stream) {
  const float* x    = (const float*)d_in[0];
  const float* W    = (const float*)d_in[1];
  const float* bias = (const float*)d_in[2];
  const float* As   = (const float*)d_in[3];
  const float* Bs   = (const float*)d_in[4];
  const float* Cs   = (const float*)d_in[5];
  float* out = (float*)d_out;

  // workspace: Weff (64 MB) then AB (2 MB)
  float* Weff = (float*)d_ws;
  float* AB   = Weff + (size_t)D_OUT * D_IN;

  ab_kernel<<<(NF * D_OUT * RANK) / 256, 256, 0, stream>>>(As, Bs, AB);
  weff_kernel<<<(D_OUT / 16) * (D_IN / 16) / 8, 256, 0, stream>>>(AB, Cs, W, Weff);
  gemm_kernel<<<dim3(D_OUT / BN, TOKENS / BM), 256, 0, stream>>>(x, Weff, bias, out);
}